// GNN_cell_18133351924122
// MI455X (gfx1250) — compile-verified
//
#include <hip/hip_runtime.h>
#include <hip/hip_bf16.h>
#include <math.h>

typedef __attribute__((ext_vector_type(2))) float v2f;
typedef __attribute__((ext_vector_type(8))) float v8f;

#define BATCH 256
#define DIM   64
#define NEGS  0.2f
#define BNEPS 1e-5f

// ---------- helpers ----------

__device__ __forceinline__ void atomicMaxF(float* addr, float value) {
    // sign-aware float max via integer atomics (correct for mixed signs, -inf init)
    if (value >= 0.0f)
        atomicMax((int*)addr, __float_as_int(value));
    else
        atomicMin((unsigned int*)addr, __float_as_uint(value));
}

__device__ __forceinline__ void edge_sd(const int* __restrict__ e, long t, int E, int N,
                                        int& src, int& dst) {
    long BE = (long)BATCH * E;
    if (t < BE) {
        int g = (int)(t / E);
        int j = (int)(t - (long)g * E);
        int off = g * N;
        src = e[j] + off;
        dst = e[E + j] + off;
    } else { // self loops appended after tiled edges
        int n = (int)(t - BE);
        src = n; dst = n;
    }
}

__device__ __forceinline__ int edge_dst(const int* __restrict__ e, long t, int E, int N) {
    long BE = (long)BATCH * E;
    if (t < BE) {
        int g = (int)(t / E);
        int j = (int)(t - (long)g * E);
        return e[E + j] + g * N;
    }
    return (int)(t - BE);
}

// ---------- kernels ----------

__global__ void fill_kernel(float* __restrict__ p, float v, long n) {
    long i = (long)blockIdx.x * blockDim.x + threadIdx.x;
    if (i < n) p[i] = v;
}

__global__ void fill4_kernel(float4* __restrict__ p, float v, long n4) {
    long i = (long)blockIdx.x * blockDim.x + threadIdx.x;
    if (i < n4) p[i] = make_float4(v, v, v, v);
}

// layer 0: [n,3] @ [3,64]
__global__ void gemm3_kernel(const float* __restrict__ x, const float* __restrict__ W,
                             float* __restrict__ h, int n_tot) {
    int i = blockIdx.x * blockDim.x + threadIdx.x;
    if (i >= n_tot * DIM) return;
    int n = i >> 6, f = i & 63;
    const float* xr = x + n * 3;
    h[i] = xr[0] * W[f] + xr[1] * W[DIM + f] + xr[2] * W[2 * DIM + f];
}

// layers 1,2: [n,64] @ [64,64] via V_WMMA_F32_16X16X4_F32
// block = 128 threads (4 waves); wave w -> column tile w; block row tile = blockIdx.x
__global__ void gemm64_wmma_kernel(const float* __restrict__ hin, const float* __restrict__ W,
                                   float* __restrict__ hout, int n_tot) {
    int wave = threadIdx.x >> 5;
    int lane = threadIdx.x & 31;
    int rowBase = blockIdx.x * 16;
    int colBase = wave * 16;
    int mrow  = rowBase + (lane & 15);
    int khalf = (lane >> 4) * 2;          // lanes 0-15: K+0/K+1 ; lanes 16-31: K+2/K+3
    int ncol  = colBase + (lane & 15);
    v8f c = {};
#pragma unroll
    for (int k0 = 0; k0 < 64; k0 += 4) {
        v2f a, b;
        a.x = hin[mrow * 64 + k0 + khalf];
        a.y = hin[mrow * 64 + k0 + khalf + 1];
        b.x = W[(k0 + khalf)     * 64 + ncol];
        b.y = W[(k0 + khalf + 1) * 64 + ncol];
        c = __builtin_amdgcn_wmma_f32_16x16x4_f32(false, a, false, b, (short)0, c,
                                                  false, false);
    }
#pragma unroll
    for (int r = 0; r < 8; ++r) {
        int row = rowBase + r + ((lane >> 4) << 3); // vgpr r: lanes 0-15 -> M=r, 16-31 -> M=r+8
        hout[row * 64 + ncol] = c[r];
    }
}

// al_s = h . a_src ; al_d = h . a_dst
__global__ void dots_kernel(const float* __restrict__ h, const float* __restrict__ asrc,
                            const float* __restrict__ adst, float* __restrict__ als,
                            float* __restrict__ ald, int n_tot) {
    int n = blockIdx.x * blockDim.x + threadIdx.x;
    if (n >= n_tot) return;
    const float4* hr = (const float4*)(h + (long)n * 64);
    const float4* a1 = (const float4*)asrc;
    const float4* a2 = (const float4*)adst;
    float s = 0.f, d = 0.f;
#pragma unroll
    for (int k = 0; k < 16; ++k) {
        float4 hv = hr[k], v1 = a1[k], v2 = a2[k];
        s += hv.x * v1.x + hv.y * v1.y + hv.z * v1.z + hv.w * v1.w;
        d += hv.x * v2.x + hv.y * v2.y + hv.z * v2.z + hv.w * v2.w;
    }
    als[n] = s; ald[n] = d;
}

// pass 1: e = leaky(al_s[src] + al_d[dst]); store e; segment max into m[dst]
__global__ void edge_logit_max_kernel(const int* __restrict__ e, const float* __restrict__ als,
                                      const float* __restrict__ ald, float* __restrict__ ebuf,
                                      float* __restrict__ m, int E, int N, long total) {
    long t = (long)blockIdx.x * blockDim.x + threadIdx.x;
    if (t >= total) return;
    int src, dst; edge_sd(e, t, E, N, src, dst);
    float v = als[src] + ald[dst];
    v = (v > 0.f) ? v : NEGS * v;
    ebuf[t] = v;
    atomicMaxF(&m[dst], v);
}

// pass 2: ex = exp(e - m[dst]); overwrite ebuf with ex; segment sum into z[dst]
__global__ void edge_exp_sum_kernel(const int* __restrict__ e, float* __restrict__ ebuf,
                                    const float* __restrict__ m, float* __restrict__ z,
                                    int E, int N, long total) {
    long t = (long)blockIdx.x * blockDim.x + threadIdx.x;
    if (t >= total) return;
    int dst = edge_dst(e, t, E, N);
    float ex = __expf(ebuf[t] - m[dst]);
    ebuf[t] = ex;
    atomicAdd(&z[dst], ex);
}

__global__ void recip_kernel(float* __restrict__ z, long n) {
    long i = (long)blockIdx.x * blockDim.x + threadIdx.x;
    if (i < n) z[i] = 1.0f / z[i];
}

// pass 3: one wave32 per edge; lane handles features 2*lane, 2*lane+1
// agg[dst][f] += (ex[t] * rz[dst]) * h[src][f]
__global__ void edge_agg_kernel(const int* __restrict__ e, const float* __restrict__ ebuf,
                                const float* __restrict__ rz, const float* __restrict__ hlin,
                                float* __restrict__ agg, int E, int N, long total) {
    long t = (long)blockIdx.x * (blockDim.x >> 5) + (threadIdx.x >> 5);
    int lane = threadIdx.x & 31;
    if (t >= total) return;
    int src, dst; edge_sd(e, t, E, N, src, dst);
    float alpha = ebuf[t] * rz[dst];
    const float2 hv = *(const float2*)(hlin + (long)src * 64 + lane * 2);
    float* ap = agg + (long)dst * 64 + lane * 2;
    atomicAdd(ap,     alpha * hv.x);
    atomicAdd(ap + 1, alpha * hv.y);
}

// bias + relu + pairwise max-pool, plus per-feature sum/sumsq partials
__global__ void pool_stats_kernel(const float* __restrict__ agg, const float* __restrict__ bias,
                                  float* __restrict__ pooled, float* __restrict__ stats,
                                  long rows_out) {
    __shared__ float ssum[64];
    __shared__ float ssq[64];
    int tid = threadIdx.x;
    if (tid < 64) { ssum[tid] = 0.f; ssq[tid] = 0.f; }
    __syncthreads();
    long i = (long)blockIdx.x * blockDim.x + tid;
    int f = (int)(i & 63);
    if (i < rows_out * 64) {
        long q = i >> 6;
        float bf = bias[f];
        float a0 = agg[(q * 2)     * 64 + f] + bf; a0 = a0 > 0.f ? a0 : 0.f;
        float a1 = agg[(q * 2 + 1) * 64 + f] + bf; a1 = a1 > 0.f ? a1 : 0.f;
        float v = a0 > a1 ? a0 : a1;
        pooled[i] = v;
        atomicAdd(&ssum[f], v);
        atomicAdd(&ssq[f], v * v);
    }
    __syncthreads();
    if (tid < 64) {
        atomicAdd(&stats[tid],      ssum[tid]);
        atomicAdd(&stats[64 + tid], ssq[tid]);
    }
}

__global__ void stats_final_kernel(const float* __restrict__ stats, float* __restrict__ mr,
                                   float inv_n) {
    int f = threadIdx.x;
    if (f < 64) {
        float mean = stats[f] * inv_n;
        float var  = stats[64 + f] * inv_n - mean * mean;
        mr[f]      = mean;
        mr[64 + f] = rsqrtf(var + BNEPS);
    }
}

__global__ void bn_kernel(const float* __restrict__ pooled, const float* __restrict__ mr,
                          float* __restrict__ out, long n) {
    long i = (long)blockIdx.x * blockDim.x + threadIdx.x;
    if (i >= n) return;
    int f = (int)(i & 63);
    out[i] = (pooled[i] - mr[f]) * mr[64 + f];
}

// ---------- host orchestration ----------

extern "C" void kernel_launch(void* const* d_in, const int* in_sizes, int n_in,
                              void* d_out, int out_size, void* d_ws, size_t ws_size,
                              hipStream_t stream) {
    const float* x     = (const float*)d_in[0];
    const float* W[3]  = {(const float*)d_in[1], (const float*)d_in[7],  (const float*)d_in[13]};
    const float* As[3] = {(const float*)d_in[2], (const float*)d_in[8],  (const float*)d_in[14]};
    const float* Ad[3] = {(const float*)d_in[3], (const float*)d_in[9],  (const float*)d_in[15]};
    const float* Bs[3] = {(const float*)d_in[4], (const float*)d_in[10], (const float*)d_in[16]};
    const int*   Ei[3] = {(const int*)d_in[5],   (const int*)d_in[11],   (const int*)d_in[17]};
    const int NSs[3]  = {4096, 2048, 1024};
    const int EPBs[3] = {32768, 16384, 8192};

    float* ws = (float*)d_ws;
    const long n0 = (long)BATCH * 4096;
    float* bufA  = ws;              // hlin / pooled           (n0*64)
    float* bufB  = bufA + n0 * 64;  // agg  / normalized out   (n0*64)
    float* als   = bufB + n0 * 64;  // (n0)
    float* ald   = als + n0;        // (n0)
    float* mbuf  = ald + n0;        // (n0)
    float* zbuf  = mbuf + n0;       // (n0)
    float* stats = zbuf + n0;       // (128) sums / sumsq
    float* mr    = stats + 128;     // (128) mean / rstd
    float* ebuf  = mr + 128;        // per-edge logit/exp (max B*32768 + n0 = 9.44M)

    const float* hin = x;
    for (int i = 0; i < 3; ++i) {
        const int  N = NSs[i], E = EPBs[i];
        const long n_tot = (long)BATCH * N;

        // 1) linear: hlin = hin @ W
        if (i == 0) {
            long tot = n_tot * 64;
            gemm3_kernel<<<(unsigned)((tot + 255) / 256), 256, 0, stream>>>(
                hin, W[0], bufA, (int)n_tot);
        } else {
            gemm64_wmma_kernel<<<(unsigned)(n_tot / 16), 128, 0, stream>>>(
                hin, W[i], bufA, (int)n_tot);
        }

        // 2) attention logits
        dots_kernel<<<(unsigned)((n_tot + 255) / 256), 256, 0, stream>>>(
            bufA, As[i], Ad[i], als, ald, (int)n_tot);

        // 3) init segment buffers
        fill_kernel<<<(unsigned)((n_tot + 255) / 256), 256, 0, stream>>>(mbuf, -INFINITY, n_tot);
        fill_kernel<<<(unsigned)((n_tot + 255) / 256), 256, 0, stream>>>(zbuf, 0.f, n_tot);
        fill4_kernel<<<(unsigned)((n_tot * 16 + 255) / 256), 256, 0, stream>>>(
            (float4*)bufB, 0.f, n_tot * 16);
        fill_kernel<<<1, 128, 0, stream>>>(stats, 0.f, 128);

        // 4) edge softmax + aggregation (edges tiled per graph + self loops)
        const long total = (long)BATCH * E + n_tot;
        edge_logit_max_kernel<<<(unsigned)((total + 255) / 256), 256, 0, stream>>>(
            Ei[i], als, ald, ebuf, mbuf, E, N, total);
        edge_exp_sum_kernel<<<(unsigned)((total + 255) / 256), 256, 0, stream>>>(
            Ei[i], ebuf, mbuf, zbuf, E, N, total);
        recip_kernel<<<(unsigned)((n_tot + 255) / 256), 256, 0, stream>>>(zbuf, n_tot);
        edge_agg_kernel<<<(unsigned)((total + 7) / 8), 256, 0, stream>>>(
            Ei[i], ebuf, zbuf, bufA, bufB, E, N, total);

        // 5) bias + relu + pair max-pool (+ BN stats), then normalize
        const long rows_out = n_tot / 2;
        pool_stats_kernel<<<(unsigned)((rows_out * 64 + 255) / 256), 256, 0, stream>>>(
            bufB, Bs[i], bufA, stats, rows_out);
        stats_final_kernel<<<1, 64, 0, stream>>>(stats, mr, 1.0f / (float)rows_out);

        float* dst = (i == 2) ? (float*)d_out : bufB;
        bn_kernel<<<(unsigned)((rows_out * 64 + 255) / 256), 256, 0, stream>>>(
            bufA, mr, dst, rows_out * 64);
        hin = dst;
    }
}